// MyModel_87522843558513
// MI455X (gfx1250) — compile-verified
//
#include <hip/hip_runtime.h>

// CDNA5 / gfx1250 GRU: WMMA F32_16X16X4_F32 recurrent GEMM, split-lane gates.
//
// Per wave: 16 sequences (WMMA columns). D(16x16) = U^T(16x8,pad) x h^T(8x16,pad) + br,
// as two chained K=4 WMMAs per GRU. Lanes 0-15 run the zero-mask GRU's elementwise
// gates; lanes 16-31 run the prev-mask GRU's gates for the same sequence column.
// The p-GRU's A/C rows are permuted by comp^8 so each lane half owns comps 0..7 of
// its own role; one select-then-shfl_xor(16) round (7 ds ops) delivers comps 8..14
// to both halves at once. Sigmoid-gate rows (comps 0..9) are pre-scaled by 0.5 in
// A, C and the gx table so each sigmoid is a bare v_tanh_f32 + fma.

typedef float v2f __attribute__((ext_vector_type(2)));
typedef float v8f __attribute__((ext_vector_type(8)));

#define VOCAB 16
#define EMB   5
#define UNITS 5
#define GDIM  15      // 3*UNITS
#define BSZ   4096
#define TLEN  1024

#if __has_builtin(__builtin_amdgcn_exp2f)
#define EXP2F(x) __builtin_amdgcn_exp2f(x)
#else
#define EXP2F(x) exp2f(x)
#endif
#if __has_builtin(__builtin_amdgcn_rcpf)
#define RCPF(x) __builtin_amdgcn_rcpf(x)
#else
#define RCPF(x) (1.0f / (x))
#endif

__device__ __forceinline__ float fast_tanh(float x) {
#if __has_builtin(__builtin_amdgcn_tanhf)
    return __builtin_amdgcn_tanhf(x);          // v_tanh_f32 (TRANS), 1 op
#else
    const float e = EXP2F(x * -2.885390081777927f);
    return (1.0f - e) * RCPF(1.0f + e);
#endif
}

// sigmoid(2*xh) given the HALF-scaled pre-activation xh (rows pre-scaled by 0.5).
__device__ __forceinline__ float sigmoid_from_half(float xh) {
#if __has_builtin(__builtin_amdgcn_tanhf)
    return fmaf(0.5f, __builtin_amdgcn_tanhf(xh), 0.5f);
#else
    return RCPF(1.0f + EXP2F(xh * -2.885390081777927f));
#endif
}

__global__ __launch_bounds__(256) void gru_dual_wmma_kernel(
    const int*   __restrict__ ids,
    const float* __restrict__ emb,
    const float* __restrict__ Wz, const float* __restrict__ Uz,
    const float* __restrict__ biz, const float* __restrict__ brz,
    const float* __restrict__ Wp, const float* __restrict__ Up,
    const float* __restrict__ bip, const float* __restrict__ brp,
    float* __restrict__ out)
{
    // G[0][id][c] = scale(c) * (bi_z[c] + (emb[id]@Wz)[c]); G[1] same for p.
    // scale(c) = 0.5 for sigmoid comps (c<10), 1.0 for candidate comps (c>=10).
    __shared__ __align__(16) float Gtab[2 * VOCAB * 16];

    const int tid = threadIdx.x;
    {
        const int id = tid >> 4;
        const int c  = tid & 15;
        float az = 0.0f, ap = 0.0f;
        if (c < GDIM) {
            az = biz[c];
            ap = bip[c];
            #pragma unroll
            for (int e = 0; e < EMB; ++e) {
                const float x = emb[id * EMB + e];
                az = fmaf(x, Wz[e * GDIM + c], az);
                ap = fmaf(x, Wp[e * GDIM + c], ap);
            }
            if (c < 10) { az *= 0.5f; ap *= 0.5f; }
        }
        Gtab[id * 16 + c]       = az;
        Gtab[256 + id * 16 + c] = ap;
    }
    __syncthreads();

    const int  lane = tid & 31;
    const bool hi   = (lane >= 16);
    const int  s    = lane & 15;                         // sequence column
    const int  wave = tid >> 5;
    const int  b    = blockIdx.x * 128 + wave * 16 + s;  // global sequence index

    // ---- A operands: U^T, K striped by lane half; sigmoid rows pre-scaled 0.5.
    //      z-GRU: row i = gate comp (lane&15).   p-GRU: row i holds comp i^8.
    v2f a1z, a2z, a1p, a2p;
    {
        const int   iz = s;
        const bool  vz = (iz < GDIM);
        const float sz = (iz < 10) ? 0.5f : 1.0f;
        a1z.x = vz ? sz * Uz[(hi ? 2 : 0) * GDIM + iz] : 0.0f;
        a1z.y = vz ? sz * Uz[(hi ? 3 : 1) * GDIM + iz] : 0.0f;
        a2z.x = (vz && !hi) ? sz * Uz[4 * GDIM + iz] : 0.0f;
        a2z.y = 0.0f;
        const int   ip = s ^ 8;                          // permuted component index
        const bool  vp = (ip < GDIM);
        const float sp = (ip < 10) ? 0.5f : 1.0f;
        a1p.x = vp ? sp * Up[(hi ? 2 : 0) * GDIM + ip] : 0.0f;
        a1p.y = vp ? sp * Up[(hi ? 3 : 1) * GDIM + ip] : 0.0f;
        a2p.x = (vp && !hi) ? sp * Up[4 * GDIM + ip] : 0.0f;
        a2p.y = 0.0f;
    }

    // ---- C operands: recurrent bias br per row (p rows ^8-permuted), same scaling.
    v8f cz, cp;
    #pragma unroll
    for (int v = 0; v < 8; ++v) {
        const int cmz  = v + (hi ? 8 : 0);
        const int cmp_ = (v + (hi ? 8 : 0)) ^ 8;
        cz[v] = (cmz  < GDIM) ? ((cmz  < 10) ? 0.5f : 1.0f) * brz[cmz]  : 0.0f;
        cp[v] = (cmp_ < GDIM) ? ((cmp_ < 10) ? 0.5f : 1.0f) * brp[cmp_] : 0.0f;
    }

    // ---- State: lo lanes carry h_z, hi lanes carry h_p (same sequence column).
    float h[UNITS] = {0, 0, 0, 0, 0};
    v2f bz1 = {0, 0}, bz2 = {0, 0}, bp1 = {0, 0}, bp2 = {0, 0};

    const size_t N1 = (size_t)BSZ * TLEN * UNITS;
    float* __restrict__ oZ = out;
    float* __restrict__ oP = out + N1;
    float* __restrict__ oE = out + 2 * N1;
    float* __restrict__ oD = out + 3 * N1;

    const float* gbase = Gtab + (hi ? 256 : 0);          // role-selected table base
    const int*   idrow = ids + (size_t)b * TLEN;
    uint32_t     o     = (uint32_t)b * (TLEN * UNITS);   // 20.97M max, fits u32

    int id = idrow[0];

    for (int t = 0; t < TLEN; ++t) {
        // Software-pipelined id load: latency overlaps this step's WMMA+gates.
        const int  id_next = idrow[(t + 1) & (TLEN - 1)];
        const bool m       = (id != 0);

        // gx row for this lane's GRU role (issues early; ds_load_b128 x4).
        const float4* grow = (const float4*)(gbase + id * 16);
        const float4 g0 = grow[0], g1 = grow[1], g2 = grow[2], g3 = grow[3];

        // gh = h @ U + br for both GRUs: two chained K=4 WMMAs each.
        v8f dz = __builtin_amdgcn_wmma_f32_16x16x4_f32(
            false, a1z, false, bz1, (short)0, cz, false, false);
        dz = __builtin_amdgcn_wmma_f32_16x16x4_f32(
            false, a2z, false, bz2, (short)0, dz, false, false);
        v8f dp = __builtin_amdgcn_wmma_f32_16x16x4_f32(
            false, a1p, false, bp1, (short)0, cp, false, false);
        dp = __builtin_amdgcn_wmma_f32_16x16x4_f32(
            false, a2p, false, bp2, (short)0, dp, false, false);

        // Layouts after the ^8 permutation on p:
        //   lo lane: dz[v] = z comp v   ; dp[v] = p comp 8+v
        //   hi lane: dz[v] = z comp 8+v ; dp[v] = p comp v
        // Select-then-shuffle: one shfl_xor(16) round serves both halves.
        const float f0 = hi ? dz[0] : dp[0];
        const float f1 = hi ? dz[1] : dp[1];
        const float f2 = hi ? dz[2] : dp[2];
        const float f3 = hi ? dz[3] : dp[3];
        const float f4 = hi ? dz[4] : dp[4];
        const float f5 = hi ? dz[5] : dp[5];
        const float f6 = hi ? dz[6] : dp[6];
        const float fx0 = __shfl_xor(f0, 16);
        const float fx1 = __shfl_xor(f1, 16);
        const float fx2 = __shfl_xor(f2, 16);
        const float fx3 = __shfl_xor(f3, 16);
        const float fx4 = __shfl_xor(f4, 16);
        const float fx5 = __shfl_xor(f5, 16);
        const float fx6 = __shfl_xor(f6, 16);

        // Role-selected gate pre-activations gh[c], c = 0..14 (comps 0..9 half-scaled).
        const float gx[GDIM] = {g0.x, g0.y, g0.z, g0.w, g1.x, g1.y, g1.z, g1.w,
                                g2.x, g2.y, g2.z, g2.w, g3.x, g3.y, g3.z};
        const float gh[GDIM] = {
            hi ? dp[0] : dz[0], hi ? dp[1] : dz[1], hi ? dp[2] : dz[2], hi ? dp[3] : dz[3],
            hi ? dp[4] : dz[4], hi ? dp[5] : dz[5], hi ? dp[6] : dz[6], hi ? dp[7] : dz[7],
            fx0, fx1, fx2, fx3, fx4, fx5, fx6};

        // Gate math: branch-free, both lane halves active (each on its own GRU).
        #pragma unroll
        for (int u = 0; u < UNITS; ++u) {
            const float zg = sigmoid_from_half(gx[u] + gh[u]);
            const float rg = sigmoid_from_half(gx[5 + u] + gh[5 + u]);
            const float cd = fast_tanh(fmaf(rg, gh[10 + u], gx[10 + u]));
            const float hn = fmaf(zg, h[u] - cd, cd);     // z*h + (1-z)*cand
            h[u] = m ? hn : h[u];                          // mask-carry state
        }

        // One set of 5 shuffles serves both the output exchange and the B rebuild:
        // lo lane receives h_p[u]; hi lane receives h_z[u].
        const float hx0 = __shfl_xor(h[0], 16);
        const float hx1 = __shfl_xor(h[1], 16);
        const float hx2 = __shfl_xor(h[2], 16);
        const float hx3 = __shfl_xor(h[3], 16);
        const float hx4 = __shfl_xor(h[4], 16);

        if (!hi) {
            const float hp[UNITS] = {hx0, hx1, hx2, hx3, hx4};
            #pragma unroll
            for (int u = 0; u < UNITS; ++u) {
                const float ozv = m ? h[u] : 0.0f;   // zero_output_for_mask
                const float opv = hp[u];             // state-carried output
                oZ[o + u] = ozv;
                oP[o + u] = opv;
                oE[o + u] = (ozv == opv) ? 1.0f : 0.0f;
                oD[o + u] = ozv - opv;
            }
        }

        // Rebuild B operands (h^T, K rows striped by lane half) for the next step.
        bz1.x = hi ? hx2 : h[0];
        bz1.y = hi ? hx3 : h[1];
        bz2.x = hi ? 0.0f : h[4];
        bz2.y = 0.0f;
        bp1.x = hi ? h[2] : hx0;
        bp1.y = hi ? h[3] : hx1;
        bp2.x = hi ? 0.0f : hx4;
        bp2.y = 0.0f;

        o += UNITS;
        id = id_next;
    }
}

extern "C" void kernel_launch(void* const* d_in, const int* in_sizes, int n_in,
                              void* d_out, int out_size, void* d_ws, size_t ws_size,
                              hipStream_t stream) {
    (void)in_sizes; (void)n_in; (void)d_ws; (void)ws_size; (void)out_size;
    const int*   ids = (const int*)d_in[0];
    const float* emb = (const float*)d_in[1];
    const float* Wz  = (const float*)d_in[2];
    const float* Uz  = (const float*)d_in[3];
    const float* biz = (const float*)d_in[4];
    const float* brz = (const float*)d_in[5];
    const float* Wp  = (const float*)d_in[6];
    const float* Up  = (const float*)d_in[7];
    const float* bip = (const float*)d_in[8];
    const float* brp = (const float*)d_in[9];

    // 16 sequences per wave, 8 waves per block -> 128 sequences/block, 32 blocks.
    dim3 grid(BSZ / 128), block(256);
    gru_dual_wmma_kernel<<<grid, block, 0, stream>>>(
        ids, emb, Wz, Uz, biz, brz, Wp, Up, bip, brp, (float*)d_out);
}